// BettiMatchingLoss_51032801411669
// MI455X (gfx1250) — compile-verified
//
#include <hip/hip_runtime.h>
#include <math.h>

// Betti-matching loss between PH0 superlevel persistence diagrams of a
// softmax-foreground prediction and a binary target, 4 samples of 64x64.
//
// Design for MI455X (gfx1250, wave32, WGP with large LDS):
//  - one workgroup per sample; the whole per-image state (values, stable
//    sort order, union-find parent array, added flags) lives in LDS (~54KB)
//  - target image is DMA'd global->LDS with GLOBAL_LOAD_ASYNC_TO_LDS_B128
//    (ASYNCcnt-tracked CDNA5 path), when the toolchain exposes the builtin
//  - stable descending argsort via O(n^2) rank counting (broadcast LDS
//    reads, deterministic, exactly matches jnp.argsort stable semantics)
//  - the union-find sweep is inherently sequential -> one lane per sample
//    runs it over LDS with path compression (roots are unchanged by
//    compression, and reference semantics only depend on roots)
//  - only persistence>0 bars can contribute to the loss, so we compact
//    them (preserving original index order => stable sort equivalence)
//  - bars + rank-sorted bars spill to d_ws; per-sample losses reduced with
//    a fixed-order LDS tree (no float atomics => deterministic replays)

#define HIMG 64
#define WIMG 64
#define NPIX (HIMG * WIMG)          // 4096
#define NT 256                      // 8 wave32 waves per workgroup
#define SAMPLES 4
#define SAMP_STRIDE 32768           // floats of ws per sample (8 arrays x 4096)
#define LOSS_OFF (SAMPLES * SAMP_STRIDE)

#if defined(__AMDGCN__) && __has_builtin(__builtin_amdgcn_global_load_async_to_lds_b128)
#define HAVE_ASYNC_LDS 1
typedef __attribute__((ext_vector_type(4))) int v4i_async;
#else
#define HAVE_ASYNC_LDS 0
#endif

__global__ __launch_bounds__(NT)
void betti_ph0_kernel(const float* __restrict__ logits,   // (4,2,64,64)
                      const float* __restrict__ tgt,      // (4,64,64)
                      float* __restrict__ ws)
{
    __shared__ __align__(16) float s_vals[NPIX + 4]; // pixel values (+ sentinel) / persistence cache
    __shared__ int            s_order[NPIX];      // stable descending order
    __shared__ int            s_parent[NPIX + 1]; // union-find parents
    __shared__ unsigned char  s_added[NPIX + 1];
    __shared__ float          s_red[NT];
    __shared__ float          s_minv;
    __shared__ int            s_m[2];             // positive-bar counts per diagram

    const int s   = blockIdx.x;
    const int tid = threadIdx.x;

    float* base  = ws + (size_t)s * SAMP_STRIDE;
    float* barBt[2] = { base,          base +  8192 };
    float* barDt[2] = { base +  4096,  base + 12288 };
    float* srtBt[2] = { base + 16384,  base + 24576 };
    float* srtDt[2] = { base + 20480,  base + 28672 };

    const float* lg0 = logits + (size_t)s * 2 * NPIX;
    const float* lg1 = lg0 + NPIX;
    const float* tg  = tgt + (size_t)s * NPIX;

    // gfx1250: lowers to global_prefetch_b8
    __builtin_prefetch(lg0 + tid * 16, 0, 0);
    __builtin_prefetch(lg1 + tid * 16, 0, 0);
    __builtin_prefetch(tg  + tid * 16, 0, 0);

    for (int t = 0; t < 2; ++t) {
        float* barB = barBt[t];
        float* barD = barDt[t];
        float* srtB = srtBt[t];
        float* srtD = srtDt[t];

        // ---- phase 1: load image values into LDS ----
        if (t == 0) {
            // jax.nn.softmax over 2 channels, take channel 1
            for (int p = tid; p < NPIX; p += NT) {
                float l0 = lg0[p], l1 = lg1[p];
                float mx = fmaxf(l0, l1);
                float e0 = expf(l0 - mx);
                float e1 = expf(l1 - mx);
                s_vals[p] = e1 / (e0 + e1);
            }
        } else {
#if HAVE_ASYNC_LDS
            // CDNA5 async DMA: global -> LDS, 16B per lane per issue,
            // tracked by ASYNCcnt (no VGPR data movement).
            for (int p4 = tid * 4; p4 < NPIX; p4 += NT * 4) {
                __builtin_amdgcn_global_load_async_to_lds_b128(
                    (v4i_async*)(tg + p4), (v4i_async*)(&s_vals[p4]), 0, 0);
            }
#if __has_builtin(__builtin_amdgcn_s_wait_asynccnt)
            __builtin_amdgcn_s_wait_asynccnt(0);
#else
            asm volatile("s_wait_asynccnt 0x0" ::: "memory");
#endif
#else
            for (int p = tid; p < NPIX; p += NT) s_vals[p] = tg[p];
#endif
        }
        if (tid == 0) {
            s_vals[NPIX]   = -__builtin_inff();  // sentinel: never added
            s_parent[NPIX] = NPIX;
            s_added[NPIX]  = 0;
        }
        __syncthreads();

        // ---- phase 2: global min of values (fixed-order tree reduce) ----
        float mn = __builtin_inff();
        for (int p = tid; p < NPIX; p += NT) mn = fminf(mn, s_vals[p]);
        s_red[tid] = mn;
        __syncthreads();
        for (int off = NT / 2; off > 0; off >>= 1) {
            if (tid < off) s_red[tid] = fminf(s_red[tid], s_red[tid + off]);
            __syncthreads();
        }
        if (tid == 0) s_minv = s_red[0];

        // ---- phase 3: stable descending argsort via rank counting ----
        // rank(p) = #{ j : v[j] > v[p]  ||  (v[j] == v[p] && j < p) }
        // (exactly jnp.argsort(-vals) stable). Also init union-find state.
        for (int p = tid; p < NPIX; p += NT) {
            const float v = s_vals[p];
            int rank = 0;
            for (int j = 0; j < NPIX; ++j) {
                const float u = s_vals[j];
                rank += (int)((u > v) || (u == v && j < p));
            }
            s_order[rank] = p;
            s_parent[p]   = p;
            s_added[p]    = 0;
        }
        __syncthreads();

        // ---- phase 4: sequential union-find sweep (single lane) ----
        if (tid == 0) {
            int m = 0;
            for (int step = 0; step < NPIX; ++step) {
                const int i = s_order[step];
                s_added[i] = 1;
                const int r = i >> 6, c = i & 63;
                int nb[4];
                nb[0] = (r > 0)        ? i - WIMG : NPIX;   // up
                nb[1] = (r < HIMG - 1) ? i + WIMG : NPIX;   // down
                nb[2] = (c > 0)        ? i - 1    : NPIX;   // left
                nb[3] = (c < WIMG - 1) ? i + 1    : NPIX;   // right
                const float vi = s_vals[i];
                for (int k = 0; k < 4; ++k) {
                    const int j = nb[k];
                    if (j == NPIX || !s_added[j]) continue;  // padding bar (zero persistence)
                    // find(i) with path compression
                    int ri = i;
                    while (s_parent[ri] != ri) ri = s_parent[ri];
                    { int x = i; while (s_parent[x] != ri) { int nx = s_parent[x]; s_parent[x] = ri; x = nx; } }
                    // find(j) with path compression
                    int rj = j;
                    while (s_parent[rj] != rj) rj = s_parent[rj];
                    { int x = j; while (s_parent[x] != rj) { int nx = s_parent[x]; s_parent[x] = rj; x = nx; } }
                    if (rj == ri) continue;                  // padding bar
                    // elder = component whose root value is higher; ties favor i's side
                    const bool i_elder = (s_vals[ri] >= s_vals[rj]);
                    const int elder = i_elder ? ri : rj;
                    const int young = i_elder ? rj : ri;
                    s_parent[young] = elder;
                    const float bb = s_vals[young];          // birth value of dying component
                    if (bb > vi) {                           // keep only persistence > 0
                        barB[m] = bb;
                        barD[m] = vi;
                        ++m;
                    }
                }
            }
            // essential bars (surviving roots die at global min); original
            // indices come after all finite bars -> append in pixel order.
            const float mv = s_minv;
            for (int p = 0; p < NPIX; ++p) {
                if (s_parent[p] == p && s_vals[p] > mv) {
                    barB[m] = s_vals[p];
                    barD[m] = mv;
                    ++m;
                }
            }
            s_m[t] = m;
        }
        __syncthreads();

        // ---- phase 5: stable sort bars by persistence descending ----
        const int m = s_m[t];
        for (int q = tid; q < m; q += NT) s_vals[q] = barB[q] - barD[q];  // reuse LDS as p-cache
        __syncthreads();
        for (int q = tid; q < m; q += NT) {
            const float pq = s_vals[q];
            int rank = 0;
            for (int j = 0; j < m; ++j) {
                const float pj = s_vals[j];
                rank += (int)((pj > pq) || (pj == pq && j < q));
            }
            srtB[rank] = barB[q];
            srtD[rank] = barD[q];
        }
        __syncthreads();
    }

    // ---- phase 6: rank-matched Betti loss ----
    const int m1 = s_m[0], m2 = s_m[1];
    const int mmin = (m1 < m2) ? m1 : m2;
    const int mmax = (m1 > m2) ? m1 : m2;
    float acc = 0.0f;
    for (int r = tid; r < mmax; r += NT) {
        if (r < mmin) {
            const float db = srtBt[0][r] - srtBt[1][r];
            const float dd = srtDt[0][r] - srtDt[1][r];
            acc += db * db + dd * dd;
        } else {
            if (r < m1) { const float d = srtBt[0][r] - srtDt[0][r]; acc += 0.5f * d * d; }
            if (r < m2) { const float d = srtBt[1][r] - srtDt[1][r]; acc += 0.5f * d * d; }
        }
    }
    s_red[tid] = acc;
    __syncthreads();
    for (int off = NT / 2; off > 0; off >>= 1) {
        if (tid < off) s_red[tid] += s_red[tid + off];
        __syncthreads();
    }
    if (tid == 0) ws[LOSS_OFF + s] = s_red[0];
}

__global__ void betti_finalize(const float* __restrict__ ws, float* __restrict__ out)
{
    if (blockIdx.x == 0 && threadIdx.x == 0) {
        float acc = 0.0f;
        for (int s = 0; s < SAMPLES; ++s) acc += ws[LOSS_OFF + s];
        out[0] = acc * (1.0f / SAMPLES);
    }
}

extern "C" void kernel_launch(void* const* d_in, const int* in_sizes, int n_in,
                              void* d_out, int out_size, void* d_ws, size_t ws_size,
                              hipStream_t stream)
{
    (void)in_sizes; (void)n_in; (void)out_size; (void)ws_size;
    const float* logits = (const float*)d_in[0];   // (4,2,64,64) f32
    const float* target = (const float*)d_in[1];   // (4,64,64)   f32
    float* ws  = (float*)d_ws;                     // needs ~513 KB
    float* out = (float*)d_out;                    // 1 float

    betti_ph0_kernel<<<SAMPLES, NT, 0, stream>>>(logits, target, ws);
    betti_finalize<<<1, 1, 0, stream>>>(ws, out);
}